// SeparableKernelWarping_41369124995308
// MI455X (gfx1250) — compile-verified
//
#include <hip/hip_runtime.h>

// Problem constants (from reference setup_inputs)
#define Bn   8
#define Hn   384
#define Wn   640
#define Cn   3
#define Kn   9
#define PADn 4
#define KLINE (2*Kn*Cn)   // 54 floats of per-pixel kernel data

// Tile config: 32x32 output tile, vertical halo of 4 on each side
#define TWn  32
#define THn  32
#define EXTn (THn + 2*PADn)   // 40 intermediate rows
#define IWn  (TWn + 2*PADn)   // 40 image cols (horizontal halo)
#define NT   256              // 8 waves (wave32)

#define GADDR __attribute__((address_space(1)))
#define LADDR __attribute__((address_space(3)))

#if defined(__has_builtin)
# if __has_builtin(__builtin_amdgcn_global_load_async_to_lds_b32)
#  define HAVE_ASYNC_LDS 1
# endif
# if __has_builtin(__builtin_amdgcn_s_wait_asynccnt)
#  define HAVE_WAIT_ASYNC_BUILTIN 1
# endif
#endif

#if defined(HAVE_ASYNC_LDS)
__device__ __forceinline__ void async_copy_f32(const float* g, float* l) {
  // (global src, lds dst, imm offset, imm cpol) -- param types confirmed by clang
  __builtin_amdgcn_global_load_async_to_lds_b32(
      (GADDR int*)g, (LADDR int*)l, 0, 0);
}
__device__ __forceinline__ void wait_async0() {
#if defined(HAVE_WAIT_ASYNC_BUILTIN)
  __builtin_amdgcn_s_wait_asynccnt(0);
#else
  asm volatile("s_wait_asynccnt 0" ::: "memory");
#endif
}
#endif

__global__ __launch_bounds__(NT)
void sepwarp_fused(const float* __restrict__ I1,
                   const float* __restrict__ I3,
                   const float* __restrict__ Ker,
                   float* __restrict__ Out)
{
  __shared__ float sA[EXTn][IWn * Cn];   // I1 tile (zero padded)
  __shared__ float sB[EXTn][IWn * Cn];   // I3 tile (zero padded)
  __shared__ float sM[EXTn][TWn * Cn];   // horizontal-pass intermediate

  const int tid = threadIdx.x;
  const int b   = blockIdx.z;
  const int r0  = blockIdx.y * THn;
  const int c0  = blockIdx.x * TWn;

  // ---------------- image tile load (rows r0-4..r0+35, cols c0-4..c0+35) ----
#if defined(HAVE_ASYNC_LDS)
  const bool interior = (r0 >= PADn) && (r0 + THn + PADn <= Hn) &&
                        (c0 >= PADn) && (c0 + TWn + PADn <= Wn);
  if (interior) {
    const size_t tile0 = ((size_t)(b * Hn + (r0 - PADn)) * Wn + (c0 - PADn)) * Cn;
    #pragma unroll 1
    for (int e = tid; e < EXTn * IWn * Cn; e += NT) {
      const int y = e / (IWn * Cn);
      const int r = e - y * (IWn * Cn);
      const size_t g = tile0 + (size_t)y * (Wn * Cn) + r;
      async_copy_f32(I1 + g, &sA[y][r]);
      async_copy_f32(I3 + g, &sB[y][r]);
    }
    wait_async0();
  } else
#endif
  {
    #pragma unroll 1
    for (int p = tid; p < EXTn * IWn; p += NT) {
      const int y  = p / IWn;
      const int x  = p - y * IWn;
      const int gy = r0 - PADn + y;
      const int gx = c0 - PADn + x;
      float a0 = 0.f, a1 = 0.f, a2 = 0.f, b0 = 0.f, b1 = 0.f, b2 = 0.f;
      if ((unsigned)gy < (unsigned)Hn && (unsigned)gx < (unsigned)Wn) {
        const size_t g = ((size_t)(b * Hn + gy) * Wn + gx) * Cn;
        a0 = I1[g]; a1 = I1[g + 1]; a2 = I1[g + 2];
        b0 = I3[g]; b1 = I3[g + 1]; b2 = I3[g + 2];
      }
      sA[y][x * 3 + 0] = a0; sA[y][x * 3 + 1] = a1; sA[y][x * 3 + 2] = a2;
      sB[y][x * 3 + 0] = b0; sB[y][x * 3 + 1] = b1; sB[y][x * 3 + 2] = b2;
    }
  }
  __syncthreads();

  // ---------------- step 1: horizontal pass into sM ------------------------
  // 40*32 = 1280 extended pixels; each wave owns one LDS row per iteration.
  #pragma unroll 1
  for (int p = tid; p < EXTn * TWn; p += NT) {
    const int y  = p / TWn;
    const int x  = p - y * TWn;
    const int gy = r0 - PADn + y;
    float a0 = 0.f, a1 = 0.f, a2 = 0.f;
    if ((unsigned)gy < (unsigned)Hn) {
      const float2* kp =
          (const float2*)(Ker + ((size_t)(b * Hn + gy) * Wn + (c0 + x)) * KLINE);
      float kb[KLINE];
      #pragma unroll
      for (int q = 0; q < KLINE / 2; ++q) {
        float2 t = kp[q];
        kb[2 * q] = t.x; kb[2 * q + 1] = t.y;
      }
      // prefetch this thread's next kernel line (dominant HBM stream)
      if (p + NT < EXTn * TWn) {
        const int pn  = p + NT;
        const int yn  = pn / TWn;
        const int xn  = pn - yn * TWn;
        const int gyn = r0 - PADn + yn;
        if ((unsigned)gyn < (unsigned)Hn)
          __builtin_prefetch(
              Ker + ((size_t)(b * Hn + gyn) * Wn + (c0 + xn)) * KLINE, 0, 3);
      }
      #pragma unroll
      for (int j = 0; j < Kn; ++j) {
        const float* iA = &sA[y][(x + j) * 3];
        const float* iB = &sB[y][(x + j) * 3];
        a0 = fmaf(iA[0], kb[j * 3 + 0], fmaf(iB[0], kb[27 + j * 3 + 0], a0));
        a1 = fmaf(iA[1], kb[j * 3 + 1], fmaf(iB[1], kb[27 + j * 3 + 1], a1));
        a2 = fmaf(iA[2], kb[j * 3 + 2], fmaf(iB[2], kb[27 + j * 3 + 2], a2));
      }
    }
    sM[y][x * 3 + 0] = a0; sM[y][x * 3 + 1] = a1; sM[y][x * 3 + 2] = a2;
  }
  __syncthreads();

  // ---------------- step 2: vertical pass with K_V = 0.5*(K1+K3) -----------
  // kernel line re-read hits L2/WGP$ (touched moments ago by step 1).
  #pragma unroll 1
  for (int p = tid; p < THn * TWn; p += NT) {
    const int y  = p / TWn;
    const int x  = p - y * TWn;
    const int gy = r0 + y;
    const int gx = c0 + x;
    const float2* kp =
        (const float2*)(Ker + ((size_t)(b * Hn + gy) * Wn + gx) * KLINE);
    float kb[KLINE];
    #pragma unroll
    for (int q = 0; q < KLINE / 2; ++q) {
      float2 t = kp[q];
      kb[2 * q] = t.x; kb[2 * q + 1] = t.y;
    }
    float a0 = 0.f, a1 = 0.f, a2 = 0.f;
    #pragma unroll
    for (int j = 0; j < Kn; ++j) {
      const float* im = &sM[y + j][x * 3];
      a0 = fmaf(im[0], 0.5f * (kb[j * 3 + 0] + kb[27 + j * 3 + 0]), a0);
      a1 = fmaf(im[1], 0.5f * (kb[j * 3 + 1] + kb[27 + j * 3 + 1]), a1);
      a2 = fmaf(im[2], 0.5f * (kb[j * 3 + 2] + kb[27 + j * 3 + 2]), a2);
    }
    // Output is write-once: nontemporal stores keep L2 free for the
    // kernels stream (whose step-2 re-read we want to hit L2, not HBM).
    float* op = Out + ((size_t)(b * Hn + gy) * Wn + gx) * Cn;
    __builtin_nontemporal_store(a0, op + 0);
    __builtin_nontemporal_store(a1, op + 1);
    __builtin_nontemporal_store(a2, op + 2);
  }
}

extern "C" void kernel_launch(void* const* d_in, const int* in_sizes, int n_in,
                              void* d_out, int out_size, void* d_ws, size_t ws_size,
                              hipStream_t stream) {
  const float* I1  = (const float*)d_in[0];
  const float* I3  = (const float*)d_in[1];
  const float* Ker = (const float*)d_in[2];
  float* Out = (float*)d_out;

  dim3 grid(Wn / TWn, Hn / THn, Bn);   // 20 x 12 x 8 = 1920 blocks
  sepwarp_fused<<<grid, dim3(NT), 0, stream>>>(I1, I3, Ker, Out);
}